// _ProposalLayer_75720273429154
// MI455X (gfx1250) — compile-verified
//
#include <hip/hip_runtime.h>
#include <stdint.h>

// ---------------- problem constants (match reference) ----------------
#define A_CNT   16320            // anchors per pyramid stack
#define B_CNT   2                // batch
#define SD      16               // temporal slots
#define TOPK    2000             // POST_NMS_TOP_N
#define N_TOT   (23 * A_CNT)     // 375360 rows per batch (W = 1 + 9 + 13)
#define NB      8192             // 13-bit radix buckets
#define CAP     4096             // candidate buffer per batch (>= TOPK)
#define HBLK    48               // histogram/gather blocks per batch

// ---------------- order-preserving float <-> uint key ----------------
__device__ __forceinline__ unsigned int fkey(float f) {
  unsigned int u = __float_as_uint(f);
  return (u & 0x80000000u) ? ~u : (u | 0x80000000u);
}
__device__ __forceinline__ float fival(unsigned int k) {
  return __uint_as_float((k & 0x80000000u) ? (k & 0x7FFFFFFFu) : ~k);
}

// scores_all[b, n]: channel 1 of the three (B,W,A,2) tensors viewed as float2.
// Levels concatenate as [W=1 | W=9 | W=13]; within a level the pair index is
// affine in n (no divisions needed): pi = b*W*A + (n - level_offset).
__device__ __forceinline__ float load_score(int b, int n,
    const float2* __restrict__ s1, const float2* __restrict__ s2,
    const float2* __restrict__ s3, int stride_ahead) {
  const float2* p;
  if (n < A_CNT) {
    p = s1 + ((size_t)b * A_CNT + n);
  } else if (n < 10 * A_CNT) {
    p = s2 + ((size_t)b * 9 * A_CNT + (n - A_CNT));
  } else {
    p = s3 + ((size_t)b * 13 * A_CNT + (n - 10 * A_CNT));
  }
  __builtin_prefetch(p + stride_ahead, 0, 0);     // global_prefetch_b8
  return p->y;
}

// ---------------- 1) bucket histogram of score keys ----------------
__global__ void hist_kernel(const float2* __restrict__ s1,
                            const float2* __restrict__ s2,
                            const float2* __restrict__ s3,
                            unsigned int* __restrict__ hist) {
  __shared__ unsigned int lh[NB];            // 32 KB LDS
  const int b = blockIdx.y;
  for (int i = threadIdx.x; i < NB; i += blockDim.x) lh[i] = 0u;
  __syncthreads();
  const int stride = gridDim.x * blockDim.x;
  for (int n = blockIdx.x * blockDim.x + threadIdx.x; n < N_TOT; n += stride) {
    float sc = load_score(b, n, s1, s2, s3, stride);
    atomicAdd(&lh[fkey(sc) >> 19], 1u);      // ds_add_u32
  }
  __syncthreads();
  for (int i = threadIdx.x; i < NB; i += blockDim.x) {
    unsigned int c = lh[i];
    if (c) atomicAdd(&hist[(size_t)b * NB + i], c);
  }
}

// ---------------- 2) find threshold bucket for rank TOPK ----------------
__global__ void scan_kernel(const unsigned int* __restrict__ hist,
                            unsigned int* __restrict__ meta) {
  __shared__ unsigned int sh[NB];            // reversed histogram copy
  __shared__ unsigned int ps[1024];          // per-thread partial sums
  const int b = blockIdx.x;
  const unsigned int* h = hist + (size_t)b * NB;
  unsigned int sum = 0;
  const int base = threadIdx.x * 8;          // reversed-index chunk [base, base+8)
  for (int k = 0; k < 8; ++k) {
    unsigned int v = h[NB - 1 - (base + k)];
    sh[base + k] = v;
    sum += v;
  }
  ps[threadIdx.x] = sum;
  __syncthreads();
  if (threadIdx.x == 0) {
    unsigned int cum = 0, above = 0;
    int T = 0; bool found = false;
    for (int c = 0; c < 1024 && !found; ++c) {
      if (cum + ps[c] >= TOPK) {
        for (int k = 0; k < 8; ++k) {
          unsigned int v = sh[c * 8 + k];
          if (cum + v >= TOPK) { above = cum; T = NB - 1 - (c * 8 + k); found = true; break; }
          cum += v;
        }
      } else {
        cum += ps[c];
      }
    }
    meta[b * 2 + 0] = (unsigned int)T;
    meta[b * 2 + 1] = above;
  }
}

// ---------------- 3) compact candidates (bucket >= T) ----------------
__global__ void gather_kernel(const float2* __restrict__ s1,
                              const float2* __restrict__ s2,
                              const float2* __restrict__ s3,
                              const unsigned int* __restrict__ meta,
                              unsigned int* __restrict__ cnt,
                              unsigned long long* __restrict__ cand) {
  const int b = blockIdx.y;
  const unsigned int T = meta[b * 2];
  const int stride = gridDim.x * blockDim.x;
  for (int n = blockIdx.x * blockDim.x + threadIdx.x; n < N_TOT; n += stride) {
    float sc = load_score(b, n, s1, s2, s3, stride);
    unsigned int k = fkey(sc);
    if ((k >> 19) >= T) {
      unsigned int pos = atomicAdd(&cnt[b], 1u);
      if (pos < CAP) {
        // descending sort on this key == top_k order (ties -> smaller index)
        cand[(size_t)b * CAP + pos] =
            (((unsigned long long)k) << 32) | (unsigned long long)(0xFFFFFFFFu - (unsigned int)n);
      }
    }
  }
}

// ---------------- 4) in-LDS bitonic sort of <=CAP candidates ----------------
__global__ void __launch_bounds__(1024)
sort_kernel(const unsigned int* __restrict__ cnt,
            const unsigned long long* __restrict__ cand,
            unsigned long long* __restrict__ top) {
  __shared__ unsigned long long sk[CAP];     // 32 KB LDS
  const int b = blockIdx.x;
  unsigned int m = cnt[b];
  if (m > CAP) m = CAP;
  // Stage candidate keys into LDS with CDNA5 async global->LDS DMA; pad with 0.
  unsigned int lds_base = (unsigned int)(uintptr_t)(&sk[0]);
  for (int i = threadIdx.x; i < CAP; i += blockDim.x) {
    if (i < (int)m) {
      unsigned int lofs = lds_base + (unsigned int)i * 8u;
      unsigned long long gaddr = (unsigned long long)(uintptr_t)(cand + (size_t)b * CAP + i);
      asm volatile("global_load_async_to_lds_b64 %0, %1, off"
                   :: "v"(lofs), "v"(gaddr) : "memory");
    } else {
      sk[i] = 0ull;                          // smallest possible key
    }
  }
  asm volatile("s_wait_asynccnt 0" ::: "memory");
  __syncthreads();
  // Bitonic sort, descending.
  for (int k2 = 2; k2 <= CAP; k2 <<= 1) {
    for (int j = k2 >> 1; j > 0; j >>= 1) {
      for (int i = threadIdx.x; i < CAP; i += blockDim.x) {
        int ixj = i ^ j;
        if (ixj > i) {
          unsigned long long a = sk[i], c = sk[ixj];
          bool desc = ((i & k2) == 0);
          if (desc ? (a < c) : (a > c)) { sk[i] = c; sk[ixj] = a; }
        }
      }
      __syncthreads();
    }
  }
  for (int r = threadIdx.x; r < TOPK; r += blockDim.x)
    top[(size_t)b * TOPK + r] = sk[r];
}

// ---------------- 5) decode + emit only the winners ----------------
__global__ void out_kernel(const unsigned long long* __restrict__ top,
                           const float* __restrict__ bb1,
                           const float* __restrict__ bb2,
                           const float* __restrict__ bb3,
                           const float* __restrict__ anchors,
                           const float* __restrict__ im_info,
                           float* __restrict__ out) {
  int t = blockIdx.x * blockDim.x + threadIdx.x;     // (b, r, s)
  if (t >= B_CNT * TOPK * SD) return;
  const int s  = t & (SD - 1);
  const int br = t >> 4;
  const int r  = br % TOPK;
  const int b  = br / TOPK;

  unsigned long long key = top[(size_t)b * TOPK + r];
  unsigned int ku = (unsigned int)(key >> 32);
  unsigned int n  = 0xFFFFFFFFu - (unsigned int)(key & 0xFFFFFFFFull);
  float score = fival(ku);

  float* row = out + (size_t)(b * TOPK + r) * 66;
  if (s == 0) { row[0] = (float)b; row[65] = score; }

  int j, a, td, Wlev; const float* bb;
  if (n < A_CNT)           { j = 0; a = (int)n; bb = bb1; Wlev = 1;  td = 16; }
  else if (n < 10 * A_CNT) { int m = (int)n - A_CNT;      j = m / A_CNT; a = m - j * A_CNT; bb = bb2; Wlev = 9;  td = 8; }
  else                     { int m = (int)n - 10 * A_CNT; j = m / A_CNT; a = m - j * A_CNT; bb = bb3; Wlev = 13; td = 4; }

  float x1, y1, x2, y2;
  if (s >= j && s < j + td) {
    float4 av = *(const float4*)(anchors + (size_t)a * 4);        // global_load_b128
    float aw  = av.z - av.x + 1.0f, ah = av.w - av.y + 1.0f;
    float acx = av.x + 0.5f * aw,   acy = av.y + 0.5f * ah;
    float4 dv = *(const float4*)(bb +
        (((size_t)(b * Wlev + j) * A_CNT + a) * td + (s - j)) * 4); // global_load_b128
    float pcx = dv.x * aw + acx, pcy = dv.y * ah + acy;
    float pw = expf(dv.z) * aw,  ph  = expf(dv.w) * ah;
    x1 = pcx - 0.5f * pw; y1 = pcy - 0.5f * ph;
    x2 = pcx + 0.5f * pw; y2 = pcy + 0.5f * ph;
  } else {
    // zero anchor + zero delta decode exactly to [0,0,1,1] in the reference
    x1 = 0.0f; y1 = 0.0f; x2 = 1.0f; y2 = 1.0f;
  }
  float hmax = im_info[b * 3 + 0] - 1.0f;
  float wmax = im_info[b * 3 + 1] - 1.0f;
  x1 = fminf(fmaxf(x1, 0.0f), wmax);
  x2 = fminf(fmaxf(x2, 0.0f), wmax);
  y1 = fminf(fmaxf(y1, 0.0f), hmax);
  y2 = fminf(fmaxf(y2, 0.0f), hmax);
  row[1 + s * 4 + 0] = x1;
  row[1 + s * 4 + 1] = y1;
  row[1 + s * 4 + 2] = x2;
  row[1 + s * 4 + 3] = y2;
}

// ---------------- host launcher ----------------
extern "C" void kernel_launch(void* const* d_in, const int* in_sizes, int n_in,
                              void* d_out, int out_size, void* d_ws, size_t ws_size,
                              hipStream_t stream) {
  (void)in_sizes; (void)n_in; (void)out_size; (void)ws_size;
  const float2* s1 = (const float2*)d_in[0];  // (B,1,A,2)
  const float2* s2 = (const float2*)d_in[1];  // (B,9,A,2)
  const float2* s3 = (const float2*)d_in[2];  // (B,13,A,2)
  const float* bb1 = (const float*)d_in[3];   // (B,1,A,16,4)
  const float* bb2 = (const float*)d_in[4];   // (B,9,A,8,4)
  const float* bb3 = (const float*)d_in[5];   // (B,13,A,4,4)
  const float* anc = (const float*)d_in[6];   // (A,4)
  const float* imi = (const float*)d_in[7];   // (B,3)
  float* out = (float*)d_out;                 // (B,TOPK,66)

  // workspace layout
  const size_t HIST_B = (size_t)B_CNT * NB * sizeof(unsigned int);  // 64 KB
  char* ws = (char*)d_ws;
  unsigned int* hist = (unsigned int*)ws;
  unsigned int* cnt  = (unsigned int*)(ws + HIST_B);                // B counters
  unsigned int* meta = (unsigned int*)(ws + HIST_B + 16);           // B*2
  unsigned long long* cand = (unsigned long long*)(ws + HIST_B + 64);
  unsigned long long* topb = cand + (size_t)B_CNT * CAP;

  // zero histogram + atomic counters (graph-capture-safe stream op)
  hipMemsetAsync(ws, 0, HIST_B + 16, stream);

  dim3 gridH(HBLK, B_CNT);
  hist_kernel<<<gridH, 256, 0, stream>>>(s1, s2, s3, hist);
  scan_kernel<<<B_CNT, 1024, 0, stream>>>(hist, meta);
  gather_kernel<<<gridH, 256, 0, stream>>>(s1, s2, s3, meta, cnt, cand);
  sort_kernel<<<B_CNT, 1024, 0, stream>>>(cnt, cand, topb);
  const int nout = B_CNT * TOPK * SD;
  out_kernel<<<(nout + 255) / 256, 256, 0, stream>>>(topb, bb1, bb2, bb3, anc, imi, out);
}